// Conv2D_33844342292804
// MI455X (gfx1250) — compile-verified
//
#include <hip/hip_runtime.h>
#include <stdint.h>

#define H 8192
#define W 8192
#define KH 7
#define KW 7
#define OH (H - KH + 1)   // 8186
#define OW (W - KW + 1)   // 8186

typedef __attribute__((ext_vector_type(2))) float v2f;
typedef __attribute__((ext_vector_type(8))) float v8f;

// 32x64 output region per block => 38x70 input halo region.
// Columns padded to 72 (K-chunk overreach reads initialized data), row stride
// padded to 76 floats: 16 rows * (76 mod 64 = 12) banks covers all 64 LDS banks
// exactly once for the per-wave ds_load_b64 A fetch -> conflict-free.
#define LDSW    76
#define LDSROWS 38
#define LDSCOLS 72

// Zero-padded banded weight table: row kh holds w[kh][0..6] at offset WTOFF,
// zeros elsewhere, so B-operand gathers need no range checks or selects.
#define WTW   40
#define WTOFF 16

__global__ __launch_bounds__(256) void conv7x7_wmma_f32(
    const float* __restrict__ x, const float* __restrict__ wgt,
    const float* __restrict__ bias, float* __restrict__ out)
{
    __shared__ float lds_x[LDSROWS * LDSW];
    __shared__ float lds_wp[KH * WTW];

    const int tid  = threadIdx.x;
    const int lane = tid & 31;
    const int wave = tid >> 5;

    const int row0 = blockIdx.y * 32;   // output region origin == input origin (VALID)
    const int col0 = blockIdx.x * 64;

    // Build the zero-padded banded weight table (zeros and weights written by
    // the same pass -> no extra barrier needed).
    for (int idx = tid; idx < KH * WTW; idx += 256) {
        int kh = idx / WTW;
        int p  = idx - kh * WTW - WTOFF;          // [-WTOFF, WTW-WTOFF)
        float v = (p >= 0 && p < KW) ? wgt[kh * KW + p] : 0.0f;
        lds_wp[idx] = v;
    }

    // Stage the 38x72 input region into LDS.
    const bool interior = (row0 + LDSROWS <= H) && (col0 + LDSCOLS <= W); // block-uniform
    if (interior) {
        // CDNA5 async copy engine: memory -> LDS b128, no VGPR round-trip,
        // tracked with ASYNCcnt. 684 16B transfers, all 16B-aligned.
        for (int idx = tid; idx < LDSROWS * (LDSCOLS / 4); idx += 256) {
            int r  = idx / (LDSCOLS / 4);
            int c4 = idx - r * (LDSCOLS / 4);
            const float* gp = x + (size_t)(row0 + r) * W + col0 + 4 * c4;
            unsigned la = (unsigned)(uintptr_t)&lds_x[r * LDSW + 4 * c4];
            uint64_t ga = (uint64_t)(uintptr_t)gp;
            asm volatile("global_load_async_to_lds_b128 %0, %1, off"
                         :: "v"(la), "v"(ga) : "memory");
        }
    } else {
        // Edge blocks: clamped scalar staging (clamping only feeds lanes whose
        // outputs are masked off, or columns killed by B-band zeros).
        for (int idx = tid; idx < LDSROWS * LDSCOLS; idx += 256) {
            int r = idx / LDSCOLS;
            int c = idx - r * LDSCOLS;
            int gr = row0 + r; gr = (gr < H) ? gr : (H - 1);
            int gc = col0 + c; gc = (gc < W) ? gc : (W - 1);
            lds_x[r * LDSW + c] = x[(size_t)gr * W + gc];
        }
    }
    asm volatile("s_wait_asynccnt 0x0" ::: "memory");
    __syncthreads();

    // wave32 WMMA operand coordinates
    const int n    = lane & 15;          // B/C/D column (N)
    const int klo  = (lane >> 4) << 1;   // K pair base: 0 for lanes 0-15, 2 for 16-31
    const int mrow = lane & 15;          // A row (M)

    const int tile_r = (wave >> 2) * 16; // 16x16 tile of the 32x64 region
    const int tile_c = (wave & 3) * 16;

    // B operands: 42 register-resident v2f band slices, pure unconditional
    // ds_loads from the zero-padded table (no clamps, no cndmasks).
    v2f Bv[KH * 6];
#pragma unroll
    for (int kh = 0; kh < KH; ++kh) {
#pragma unroll
        for (int c = 0; c < 6; ++c) {
            const float* wp = &lds_wp[kh * WTW + WTOFF + (4 * c + klo - n)]; // idx 1..39
            v2f bv;
            bv.x = wp[0];
            bv.y = wp[1];
            Bv[kh * 6 + c] = bv;
        }
    }

    // Main accumulation: 42 x v_wmma_f32_16x16x4_f32 into one 16x16 f32 tile.
    v8f acc = {};
#pragma unroll
    for (int kh = 0; kh < KH; ++kh) {
        const int ldrow = tile_r + mrow + kh;        // <= 37 < 38
#pragma unroll
        for (int c = 0; c < 6; ++c) {
            const int ldcol = tile_c + 4 * c + klo;  // <= 70 (+1 for .y = 71)
            v2f a = *(const v2f*)&lds_x[ldrow * LDSW + ldcol];
            acc = __builtin_amdgcn_wmma_f32_16x16x4_f32(
                /*neg_a=*/false, a, /*neg_b=*/false, Bv[kh * 6 + c],
                /*c_mod=*/(short)0, acc, /*reuse_a=*/false, /*reuse_b=*/false);
        }
    }

    // Store: D VGPR v -> M = v + 8*(lane>=16), N = lane&15.
    const float bval = bias[0];
    const int mbase  = (lane >> 4) << 3;
#pragma unroll
    for (int v = 0; v < 8; ++v) {
        int orow = row0 + tile_r + mbase + v;
        int ocol = col0 + tile_c + n;
        if (orow < OH && ocol < OW) {
            out[(size_t)orow * OW + ocol] = acc[v] + bval;
        }
    }
}

extern "C" void kernel_launch(void* const* d_in, const int* in_sizes, int n_in,
                              void* d_out, int out_size, void* d_ws, size_t ws_size,
                              hipStream_t stream) {
    const float* x    = (const float*)d_in[0];
    const float* wgt  = (const float*)d_in[1];
    const float* bias = (const float*)d_in[2];
    float* out        = (float*)d_out;

    dim3 grid((OW + 63) / 64, (OH + 31) / 32);  // 128 x 256 blocks
    conv7x7_wmma_f32<<<grid, 256, 0, stream>>>(x, wgt, bias, out);
}